// WaveCell_13838384627861
// MI455X (gfx1250) — compile-verified
//
#include <hip/hip_runtime.h>
#include <hip/hip_bf16.h>

#define DT      0.1f
#define DT_HALF 0.05f
#define DT_SQ   0.01f

#define NN   256      // nodes
#define FF   512      // features
#define BB   64       // batch
#define NFX  16       // fixed points
#define AUGW 768      // augmented width: [S | 2M-dt^2 K | M]

typedef float v2f __attribute__((ext_vector_type(2)));
typedef float v8f __attribute__((ext_vector_type(8)));

// ---------------------------------------------------------------------------
// 1) Build augmented system  Aug[i] = [ S | R1 | M ]  with
//    S = M + dt/2*C,  R1 = 2M - dt^2*K   (graph-Laplacian assembly inline)
// ---------------------------------------------------------------------------
__global__ __launch_bounds__(NN) void build_aug_kernel(
    const float* __restrict__ kn, const float* __restrict__ kc,
    const float* __restrict__ mn, const float* __restrict__ mc,
    const float* __restrict__ cn, const float* __restrict__ cc,
    const float* __restrict__ coup, float* __restrict__ Aug) {
  const int i = blockIdx.x;
  const int j = threadIdx.x;
  const int ij = i * NN + j;

  const float cp = coup[ij];
  const float eK = kc[ij] * cp;
  const float eM = mc[ij] * cp;
  const float eC = cc[ij] * cp;

  __shared__ float sK[NN], sM[NN], sC[NN];
  sK[j] = eK; sM[j] = eM; sC[j] = eC;
  __syncthreads();
  for (int s = NN / 2; s > 0; s >>= 1) {
    if (j < s) { sK[j] += sK[j + s]; sM[j] += sM[j + s]; sC[j] += sC[j + s]; }
    __syncthreads();
  }
  const float rsK = sK[0], rsM = sM[0], rsC = sC[0];

  const float Kij = -eK + ((i == j) ? (kn[i] + rsK) : 0.0f);
  const float Mij = -eM + ((i == j) ? (mn[i] + rsM) : 0.0f);
  const float Cij = -eC + ((i == j) ? (cn[i] + rsC) : 0.0f);

  float* row = Aug + (long)i * AUGW;
  row[j]            = Mij + DT_HALF * Cij;        // S
  row[NN + j]       = 2.0f * Mij - DT_SQ * Kij;   // R1
  row[2 * NN + j]   = Mij;                        // M (RHS2)
}

// ---------------------------------------------------------------------------
// 2) Gauss-Jordan (no pivoting; S strictly diagonally dominant).
//    Single workgroup, 1024 threads; pivot row staged in LDS.
//    Result: Aug[:,256:512] = inv(S)(2M-dt^2 K),  Aug[:,512:768] = inv(S) M
// ---------------------------------------------------------------------------
__global__ __launch_bounds__(1024) void gauss_jordan_kernel(float* __restrict__ Aug) {
  const int tid = threadIdx.x;
  const int wv = tid >> 5, ln = tid & 31;
  __shared__ float prow[AUGW];
  __shared__ float fac[NN];

  for (int p = 0; p < NN; ++p) {
    float* rowp = Aug + (long)p * AUGW;
    const float piv = rowp[p];
    __syncthreads();                       // everyone read pivot before overwrite
    const float rp = 1.0f / piv;
    for (int j = tid; j < AUGW; j += 1024) {
      const float v = rowp[j] * rp;
      rowp[j] = v;
      prow[j] = v;
    }
    __syncthreads();
    if (tid < NN) fac[tid] = (tid == p) ? 0.0f : Aug[(long)tid * AUGW + p];
    __syncthreads();
    for (int r = wv; r < NN; r += 32) {    // one wave per row, 8 rows/wave
      const float f = fac[r];
      float* rowr = Aug + (long)r * AUGW;
      for (int j = ln; j < AUGW; j += 32) rowr[j] -= f * prow[j];
    }
    __syncthreads();
  }
}

// ---------------------------------------------------------------------------
// 3) Finalize: write T1, T2 TRANSPOSED (Tt[k][i]) so WMMA A-fragment loads
//    are lane-contiguous.  T1 = X1,  T2 = I - 2*X2.
// ---------------------------------------------------------------------------
__global__ __launch_bounds__(NN) void finalize_T_kernel(
    const float* __restrict__ Aug, float* __restrict__ T1t, float* __restrict__ T2t) {
  const int i = blockIdx.x;   // row of T
  const int j = threadIdx.x;  // col of T
  const float x1 = Aug[(long)i * AUGW + NN + j];
  const float x2 = Aug[(long)i * AUGW + 2 * NN + j];
  T1t[j * NN + i] = x1;
  T2t[j * NN + i] = ((i == j) ? 1.0f : 0.0f) - 2.0f * x2;
}

// ---------------------------------------------------------------------------
// 4) Mask h1 rows listed in fxp -> h1m (second half of d_out)
// ---------------------------------------------------------------------------
__global__ __launch_bounds__(256) void mask_h1_kernel(
    const float* __restrict__ h1, const int* __restrict__ fxp,
    float* __restrict__ h1m) {
  __shared__ int sfx[NFX];
  if (threadIdx.x < NFX) sfx[threadIdx.x] = fxp[threadIdx.x];
  __syncthreads();
  const long idx = (long)blockIdx.x * blockDim.x + threadIdx.x;
  const int row = (int)((idx >> 9) & (NN - 1));   // (idx / FF) % NN
  bool zero = false;
#pragma unroll
  for (int t = 0; t < NFX; ++t) zero |= (sfx[t] == row);
  h1m[idx] = zero ? 0.0f : h1[idx];
}

// ---------------------------------------------------------------------------
// 5) Batched GEMM via fp32 WMMA:  Y_b = T1 @ H1m_b + T2 @ H2_b
//    grid = (F/128, N/64, B); block = 256 (8 waves).
//    Wave w -> F-stripe of 16 cols; 4 accumulators cover 64 rows.
//    A-frag (16x4 f32): VGPR0 = K+2*half, VGPR1 = K+2*half+1, lanes = M rows.
//    B-frag (4x16 f32): same K split, lanes = N cols.
// ---------------------------------------------------------------------------
__global__ __launch_bounds__(256) void wmma_gemm_kernel(
    const float* __restrict__ T1t, const float* __restrict__ T2t,
    const float* __restrict__ h1m, const float* __restrict__ h2,
    float* __restrict__ y) {
  const int lane = threadIdx.x & 31;
  const int wave = threadIdx.x >> 5;       // 0..7
  const int half = lane >> 4;              // 0/1
  const int l    = lane & 15;

  const int f0 = blockIdx.x * 128 + wave * 16;
  const int m0 = blockIdx.y * 64;
  const long hb = (long)blockIdx.z * NN * FF;

  const float* __restrict__ H1 = h1m + hb;
  const float* __restrict__ H2 = h2 + hb;

  v8f acc[4] = {};

  for (int k = 0; k < NN; k += 4) {
    const int kk = k + 2 * half;
    // B fragments (shared across the 4 row tiles)
    v2f b1, b2;
    b1.x = H1[(long)kk * FF + f0 + l];
    b1.y = H1[(long)(kk + 1) * FF + f0 + l];
    b2.x = H2[(long)kk * FF + f0 + l];
    b2.y = H2[(long)(kk + 1) * FF + f0 + l];
#pragma unroll
    for (int t = 0; t < 4; ++t) {
      const int r = m0 + t * 16 + l;       // M row for this lane
      v2f a1, a2;
      a1.x = T1t[kk * NN + r];             // lane-contiguous (T stored transposed)
      a1.y = T1t[(kk + 1) * NN + r];
      a2.x = T2t[kk * NN + r];
      a2.y = T2t[(kk + 1) * NN + r];
      acc[t] = __builtin_amdgcn_wmma_f32_16x16x4_f32(
          false, a1, false, b1, (short)0, acc[t], false, false);
      acc[t] = __builtin_amdgcn_wmma_f32_16x16x4_f32(
          false, a2, false, b2, (short)0, acc[t], false, false);
    }
  }

  float* __restrict__ Y = y + hb;
#pragma unroll
  for (int t = 0; t < 4; ++t) {
#pragma unroll
    for (int v = 0; v < 8; ++v) {
      const int row = m0 + t * 16 + v + 8 * half;  // C/D layout: VGPR v, half
      Y[(long)row * FF + f0 + l] = acc[t][v];
    }
  }
}

// ---------------------------------------------------------------------------
extern "C" void kernel_launch(void* const* d_in, const int* in_sizes, int n_in,
                              void* d_out, int out_size, void* d_ws, size_t ws_size,
                              hipStream_t stream) {
  const float* kn   = (const float*)d_in[0];
  const float* kc   = (const float*)d_in[1];
  const float* mn   = (const float*)d_in[2];
  const float* mc   = (const float*)d_in[3];
  const float* cn   = (const float*)d_in[4];
  const float* cc   = (const float*)d_in[5];
  const float* h1   = (const float*)d_in[6];
  const float* h2   = (const float*)d_in[7];
  const float* coup = (const float*)d_in[8];
  const int*   fxp  = (const int*)d_in[9];

  float* y   = (float*)d_out;                         // [B,N,F]
  float* h1m = y + (long)BB * NN * FF;                // second output [B,N,F]

  float* Aug = (float*)d_ws;                          // 256*768 floats
  float* T1t = Aug + NN * AUGW;                       // 256*256
  float* T2t = T1t + NN * NN;                         // 256*256

  // Small dense preamble
  build_aug_kernel<<<NN, NN, 0, stream>>>(kn, kc, mn, mc, cn, cc, coup, Aug);
  gauss_jordan_kernel<<<1, 1024, 0, stream>>>(Aug);
  finalize_T_kernel<<<NN, NN, 0, stream>>>(Aug, T1t, T2t);

  // h1m output (also GEMM input)
  const long nh = (long)BB * NN * FF;                 // 8,388,608
  mask_h1_kernel<<<(int)(nh / 256), 256, 0, stream>>>(h1, fxp, h1m);

  // Heavy batched GEMM on the WMMA path
  dim3 grid(FF / 128, NN / 64, BB);
  wmma_gemm_kernel<<<grid, 256, 0, stream>>>(T1t, T2t, h1m, h2, y);
}